// SSMLayer_14345190768767
// MI455X (gfx1250) — compile-verified
//
#include <hip/hip_runtime.h>
#include <hip/hip_bf16.h>

// SSM layer: BATCH=8, SEQ=4096, DIM=1024, STATE=64
// y = C @ scan(a, u @ (delta*B)^T) + D*u   (fp32; memory-bound ~20 FLOP/B ->
// fp32 WMMA 16x16x4 + TDM tensor_load_to_lds for B-operand staging)

#define BATCH 8
#define SEQ   4096
#define DIM   1024
#define NST   64
#define ROWS  (BATCH*SEQ)      // 32768
#define NCOLS (BATCH*NST)      // 512 series for the scan
#define CHLEN 128              // scan chunk length
#define NCHUNK (SEQ/CHLEN)     // 32

typedef __attribute__((ext_vector_type(2))) float v2f;
typedef __attribute__((ext_vector_type(8))) float v8f;
typedef __attribute__((ext_vector_type(4))) unsigned int u32x4;
typedef __attribute__((ext_vector_type(8))) unsigned int u32x8;

// ---------------------------------------------------------------------------
// TDM: DMA a 2D fp32 tile (tile_d0 x tile_d1 dwords, row stride stride_d0
// dwords in global) into LDS at lds_byte, inserting pad_amt(+1) dwords of LDS
// padding every 2^(pad_int+1) dwords (bank-conflict-free row stride for the
// WMMA B-fragment ds_load_b64s). D# layout per CDNA5 ISA.
// ---------------------------------------------------------------------------
static __device__ __forceinline__ void tdm_load_2d(unsigned lds_byte,
                                                   unsigned long long gaddr,
                                                   unsigned tensor_d0, unsigned tensor_d1,
                                                   unsigned tile_d0, unsigned tile_d1,
                                                   unsigned long long stride_d0,
                                                   unsigned pad_int_code, unsigned pad_amt_code) {
    u32x4 g0;
    g0[0] = 1u;                                                    // count=1, user mode
    g0[1] = lds_byte;                                              // lds_addr
    g0[2] = (unsigned)gaddr;                                       // global_addr[31:0]
    g0[3] = ((unsigned)(gaddr >> 32) & 0x01FFFFFFu) | (2u << 30);  // addr[56:32] | type=2
    u32x8 g1;
    g1[0] = (2u << 16)                    // data_size = 4 bytes
          | (1u << 20)                    // pad_enable
          | ((pad_int_code & 7u) << 22)   // pad every 2^(code+1) dwords
          | ((pad_amt_code & 127u) << 25);// pad amount = code+1 dwords
    g1[1] = (tensor_d0 & 0xFFFFu) << 16;                           // dim0[15:0] @bits63:48
    g1[2] = ((tensor_d0 >> 16) & 0xFFFFu) | ((tensor_d1 & 0xFFFFu) << 16);
    g1[3] = ((tensor_d1 >> 16) & 0xFFFFu) | ((tile_d0 & 0xFFFFu) << 16);
    g1[4] = tile_d1 & 0xFFFFu;                                     // tile_dim1 (tile_dim2=0)
    g1[5] = (unsigned)stride_d0;                                   // dim0_stride[31:0]
    g1[6] = (unsigned)((stride_d0 >> 32) & 0xFFFFu);               // dim0_stride[47:32]
    g1[7] = 0u;
    asm volatile("tensor_load_to_lds %0, %1" :: "s"(g0), "s"(g1) : "memory");
}

// ---------------------------------------------------------------------------
// prep: w[n][d] = softplus(log_delta[d]) * B[n][d]
//       a[n]   = mean_d exp(softplus(log_delta[d]) * (-exp(log_A_real[n])))
// ---------------------------------------------------------------------------
__global__ void ssm_prep(const float* __restrict__ logA,
                         const float* __restrict__ Bm,
                         const float* __restrict__ logdelta,
                         float* __restrict__ w,
                         float* __restrict__ a_out) {
    __shared__ float red[256];
    const int n = blockIdx.x;
    const int tid = threadIdx.x;
    const float Ar = -__expf(logA[n]);
    float partial = 0.f;
    for (int d = tid; d < DIM; d += 256) {
        float x = logdelta[d];
        float delta = (x > 20.f) ? x : log1pf(__expf(x));   // softplus
        w[n * DIM + d] = delta * Bm[n * DIM + d];
        partial += __expf(delta * Ar);
    }
    red[tid] = partial;
    __syncthreads();
    for (int s = 128; s > 0; s >>= 1) {
        if (tid < s) red[tid] += red[tid + s];
        __syncthreads();
    }
    if (tid == 0) a_out[n] = red[0] * (1.0f / (float)DIM);
}

// ---------------------------------------------------------------------------
// gemm1: Bu[l*8+b][n] = sum_d u[b][l][d] * w[n][d]
// M=32768, N=64, K=1024. WMMA f32 16x16x4, wave tile 16x64.
// w staged by the TDM in K-chunks of 128, double-buffered; TDM pad gives LDS
// row stride 132 dwords (lane bank step 4 => conflict-free b64 reads).
// A fragment (16x4 fp32): lanes 0-15 hold (M=lane, K={0,1}),
//                         lanes 16-31 hold (M=lane-16, K={2,3}).
// ---------------------------------------------------------------------------
#define KCH 128
#define WSTR 132
__global__ void ssm_gemm1(const float* __restrict__ u,
                          const float* __restrict__ w,
                          float* __restrict__ Bu) {
    __shared__ float lw[2 * NST * WSTR];
    const int tid  = threadIdx.x;
    const int lane = tid & 31;
    const int wave = tid >> 5;
    const int m    = lane & 15;
    const int khalf = (lane >> 4) << 1;   // 0 or 2
    const int rowbase = blockIdx.x * 128 + wave * 16;
    const int r = rowbase + m;            // Bu row (l*8+b)
    const int l = r >> 3, b = r & 7;
    const float* urow = u + ((size_t)(b * SEQ + l)) * DIM;

    const unsigned lds0 = (unsigned)(size_t)&lw[0];
    const unsigned lds1 = (unsigned)(size_t)&lw[NST * WSTR];

    // prologue: DMA chunk 0 into buffer 0
    if (wave == 0) {
        tdm_load_2d(lds0, (unsigned long long)(size_t)w,
                    KCH, NST, KCH, NST, DIM, /*pad: every 128 dw*/6u, /*+4 dw*/3u);
        __builtin_amdgcn_s_wait_tensorcnt(0);
    }
    __syncthreads();

    v8f acc0 = {}, acc1 = {}, acc2 = {}, acc3 = {};
    int cur = 0;
    for (int kb = 0; kb < DIM; kb += KCH) {
        // overlap: DMA next chunk into the other buffer while we compute
        if (wave == 0 && (kb + KCH) < DIM) {
            tdm_load_2d(cur ? lds0 : lds1,
                        (unsigned long long)(size_t)w + (unsigned long long)(kb + KCH) * 4ull,
                        KCH, NST, KCH, NST, DIM, 6u, 3u);
        }
        const float* lwc = &lw[cur * (NST * WSTR)];
        #pragma unroll 4
        for (int kk = 0; kk < KCH; kk += 4) {
            v2f af = *(const v2f*)(urow + kb + kk + khalf);
            v2f b0 = *(const v2f*)(&lwc[(0 * 16 + m) * WSTR + kk + khalf]);
            v2f b1 = *(const v2f*)(&lwc[(1 * 16 + m) * WSTR + kk + khalf]);
            v2f b2 = *(const v2f*)(&lwc[(2 * 16 + m) * WSTR + kk + khalf]);
            v2f b3 = *(const v2f*)(&lwc[(3 * 16 + m) * WSTR + kk + khalf]);
            acc0 = __builtin_amdgcn_wmma_f32_16x16x4_f32(false, af, false, b0, (short)0, acc0, false, false);
            acc1 = __builtin_amdgcn_wmma_f32_16x16x4_f32(false, af, false, b1, (short)0, acc1, false, false);
            acc2 = __builtin_amdgcn_wmma_f32_16x16x4_f32(false, af, false, b2, (short)0, acc2, false, false);
            acc3 = __builtin_amdgcn_wmma_f32_16x16x4_f32(false, af, false, b3, (short)0, acc3, false, false);
        }
        if (wave == 0) __builtin_amdgcn_s_wait_tensorcnt(0);  // next buffer landed
        __syncthreads();
        cur ^= 1;
    }

    // C/D layout: VGPR j -> M=j (lanes 0-15) or M=j+8 (lanes 16-31), N=lane&15
    const int khi = lane >> 4;
    #pragma unroll
    for (int j = 0; j < 8; ++j) {
        int orow = rowbase + j + (khi << 3);
        float* op = Bu + (size_t)orow * NST + m;
        op[0]  = acc0[j];
        op[16] = acc1[j];
        op[32] = acc2[j];
        op[48] = acc3[j];
    }
}

// ---------------------------------------------------------------------------
// Chunked parallel scan over l (series = (b,n), 512 of them, a[n] scalar decay)
// ---------------------------------------------------------------------------
__global__ void ssm_scan_carry(const float* __restrict__ Bu,
                               const float* __restrict__ a,
                               float* __restrict__ carry) {
    int gid = blockIdx.x * 256 + threadIdx.x;
    int c = gid >> 9, j = gid & 511;
    float av = a[j & 63];
    const float* p = Bu + (size_t)c * CHLEN * NCOLS + j;
    float s = 0.f;
    #pragma unroll 8
    for (int t = 0; t < CHLEN; ++t) s = fmaf(s, av, p[(size_t)t * NCOLS]);
    carry[c * NCOLS + j] = s;
}

__global__ void ssm_scan_combine(const float* __restrict__ carry,
                                 const float* __restrict__ a,
                                 float* __restrict__ cin) {
    int j = threadIdx.x;
    float av = a[j & 63];
    float aL = av;
    #pragma unroll
    for (int i = 0; i < 7; ++i) aL *= aL;   // a^128
    float s = 0.f;
    for (int c = 0; c < NCHUNK; ++c) {
        cin[c * NCOLS + j] = s;             // state entering chunk c
        s = fmaf(s, aL, carry[c * NCOLS + j]);
    }
}

__global__ void ssm_scan_apply(float* __restrict__ Bu,   // in-place -> xs
                               const float* __restrict__ a,
                               const float* __restrict__ cin) {
    int gid = blockIdx.x * 256 + threadIdx.x;
    int c = gid >> 9, j = gid & 511;
    float av = a[j & 63];
    float* p = Bu + (size_t)c * CHLEN * NCOLS + j;
    float s = cin[c * NCOLS + j];
    for (int t = 0; t < CHLEN; ++t) {
        s = fmaf(s, av, p[(size_t)t * NCOLS]);
        p[(size_t)t * NCOLS] = s;
    }
}

// ---------------------------------------------------------------------------
// gemm2 + epilogue: y[r][d] = sum_n C[d][n]*xs[lb(r)][n] + D[d]*u[r][d]
// M=32768 (y-order rows), N=1024, K=64. C tile (64x64) DMA'd by the TDM with
// pad -> LDS row stride 68 dwords (conflict-free b64 B-fragment reads).
// ---------------------------------------------------------------------------
#define CSTR 68
__global__ void ssm_gemm2(const float* __restrict__ xs,
                          const float* __restrict__ Cm,
                          const float* __restrict__ u,
                          const float* __restrict__ Dv,
                          float* __restrict__ y) {
    __shared__ float lc[NST * CSTR];
    const int tid  = threadIdx.x;
    const int lane = tid & 31;
    const int wave = tid >> 5;
    const int m    = lane & 15;
    const int khalf = (lane >> 4) << 1;
    const int dbase = blockIdx.y * 64;

    // stage C[dbase..dbase+64)[0..64) row-major over k via TDM
    if (wave == 0) {
        tdm_load_2d((unsigned)(size_t)&lc[0],
                    (unsigned long long)(size_t)(Cm + (size_t)dbase * NST),
                    NST, 64u, NST, 64u, NST, /*pad: every 64 dw*/5u, /*+4 dw*/3u);
        __builtin_amdgcn_s_wait_tensorcnt(0);
    }
    __syncthreads();

    const int rowbase = blockIdx.x * 128 + wave * 16;
    const int r = rowbase + m;          // y-order row: r = b*SEQ + l
    const int b = r >> 12, l = r & (SEQ - 1);
    const float* xrow = xs + (size_t)(l * 8 + b) * NST;

    v8f acc0 = {}, acc1 = {}, acc2 = {}, acc3 = {};
    #pragma unroll
    for (int kk = 0; kk < NST; kk += 4) {
        v2f af = *(const v2f*)(xrow + kk + khalf);
        v2f b0 = *(const v2f*)(&lc[(0 * 16 + m) * CSTR + kk + khalf]);
        v2f b1 = *(const v2f*)(&lc[(1 * 16 + m) * CSTR + kk + khalf]);
        v2f b2 = *(const v2f*)(&lc[(2 * 16 + m) * CSTR + kk + khalf]);
        v2f b3 = *(const v2f*)(&lc[(3 * 16 + m) * CSTR + kk + khalf]);
        acc0 = __builtin_amdgcn_wmma_f32_16x16x4_f32(false, af, false, b0, (short)0, acc0, false, false);
        acc1 = __builtin_amdgcn_wmma_f32_16x16x4_f32(false, af, false, b1, (short)0, acc1, false, false);
        acc2 = __builtin_amdgcn_wmma_f32_16x16x4_f32(false, af, false, b2, (short)0, acc2, false, false);
        acc3 = __builtin_amdgcn_wmma_f32_16x16x4_f32(false, af, false, b3, (short)0, acc3, false, false);
    }

    const int khi = lane >> 4;
    const float d0 = Dv[dbase + 0  + m];
    const float d1 = Dv[dbase + 16 + m];
    const float d2 = Dv[dbase + 32 + m];
    const float d3 = Dv[dbase + 48 + m];
    #pragma unroll
    for (int j = 0; j < 8; ++j) {
        int orow = rowbase + j + (khi << 3);
        const float* up = u + (size_t)orow * DIM + dbase + m;
        float*       op = y + (size_t)orow * DIM + dbase + m;
        op[0]  = fmaf(d0, up[0],  acc0[j]);
        op[16] = fmaf(d1, up[16], acc1[j]);
        op[32] = fmaf(d2, up[32], acc2[j]);
        op[48] = fmaf(d3, up[48], acc3[j]);
    }
}

// ---------------------------------------------------------------------------
extern "C" void kernel_launch(void* const* d_in, const int* in_sizes, int n_in,
                              void* d_out, int out_size, void* d_ws, size_t ws_size,
                              hipStream_t stream) {
    (void)in_sizes; (void)n_in; (void)out_size; (void)ws_size;
    const float* u        = (const float*)d_in[0];
    const float* logA     = (const float*)d_in[1];
    const float* Bm       = (const float*)d_in[2];
    const float* Cm       = (const float*)d_in[3];
    const float* Dv       = (const float*)d_in[4];
    const float* logdelta = (const float*)d_in[5];
    float* y  = (float*)d_out;
    float* ws = (float*)d_ws;

    float* w     = ws;                          // 64*1024
    float* a     = w + NST * DIM;               // 64 (+pad)
    float* Bu    = a + 128;                     // 32768*64  (becomes xs in place)
    float* carry = Bu + (size_t)ROWS * NST;     // 32*512
    float* cin   = carry + NCHUNK * NCOLS;      // 32*512

    ssm_prep<<<NST, 256, 0, stream>>>(logA, Bm, logdelta, w, a);
    ssm_gemm1<<<ROWS / 128, 256, 0, stream>>>(u, w, Bu);
    ssm_scan_carry<<<(NCHUNK * NCOLS) / 256, 256, 0, stream>>>(Bu, a, carry);
    ssm_scan_combine<<<1, NCOLS, 0, stream>>>(carry, a, cin);
    ssm_scan_apply<<<(NCHUNK * NCOLS) / 256, 256, 0, stream>>>(Bu, a, cin);
    ssm_gemm2<<<dim3(ROWS / 128, DIM / 64), 256, 0, stream>>>(Bu, Cm, u, Dv, y);
}